// GATHead_90847148245496
// MI455X (gfx1250) — compile-verified
//
#include <hip/hip_runtime.h>
#include <math.h>

// ---------------------------------------------------------------------------
// GAT (2 layers) on MI455X / gfx1250, wave32, WMMA f32 16x16x4.
// Pipeline: CSR build -> WMMA GEMM1 -> el/er -> per-node online-softmax
// aggregation -> WMMA GEMM2 -> el/er -> aggregation -> LDS-tiled transpose.
// Problem is memory-bound (~1.5 GFLOP vs ~50MB traffic), so f32 WMMA keeps
// reference precision at no cost; all gathers are coalesced feature rows.
// ---------------------------------------------------------------------------

#define NNODES 4096
#define BATCHN 16
#define INCH   128
#define HID1   32      // HEADS*HID = 4*8
#define OUTCH  64

typedef float v2f __attribute__((ext_vector_type(2)));
typedef float v8f __attribute__((ext_vector_type(8)));

// ------------------------------- CSR build ---------------------------------
__global__ void k_zero(int* counts, int* cursor, int n) {
  int i = blockIdx.x * blockDim.x + threadIdx.x;
  if (i < n) { counts[i] = 0; cursor[i] = 0; }
}

__global__ void k_count(const int* __restrict__ dst, int* counts, int E) {
  int e = blockIdx.x * blockDim.x + threadIdx.x;
  if (e < E) atomicAdd(&counts[dst[e]], 1);
}

__global__ void k_scan(const int* __restrict__ counts, int* offs) {
  if (blockIdx.x == 0 && threadIdx.x == 0) {
    int acc = 0;
    for (int i = 0; i < NNODES; ++i) { offs[i] = acc; acc += counts[i]; }
    offs[NNODES] = acc;
  }
}

__global__ void k_fill(const int* __restrict__ src, const int* __restrict__ dst,
                       const int* __restrict__ offs, int* cursor,
                       int* __restrict__ esrc, int E) {
  int e = blockIdx.x * blockDim.x + threadIdx.x;
  if (e < E) {
    int d = dst[e];
    int p = atomicAdd(&cursor[d], 1);
    esrc[offs[d] + p] = src[e];
  }
}

// ------------------------- GEMM1: feat1 = x^T @ W1 -------------------------
// x: [B][128][4096]  W1: [128][32]  feat1: [B][4096][32]
// block = 4 waves; each wave owns one 16x16 output tile; W1 slab staged in LDS
__global__ void k_gemm1(const float* __restrict__ x, const float* __restrict__ W1,
                        float* __restrict__ feat1) {
  __shared__ float Bsh[INCH * 16];                 // 128K x 16N slab, 8KB
  const int lane = threadIdx.x & 31;
  const int wv   = threadIdx.x >> 5;               // 0..3
  const int b    = blockIdx.z;
  const int j0   = blockIdx.y * 16;                // output-channel tile
  const int m0   = (blockIdx.x * 4 + wv) * 16;     // node-row tile
  const int m    = lane & 15;
  const int kh   = lane >> 4;                      // K half select (ISA A layout)

  for (int idx = threadIdx.x; idx < INCH * 16; idx += blockDim.x) {
    int k = idx >> 4, nn = idx & 15;
    Bsh[idx] = W1[k * HID1 + j0 + nn];
  }
  __syncthreads();

  const float* xb = x + (size_t)b * INCH * NNODES;
  v8f acc = {};
  for (int k0 = 0; k0 < INCH; k0 += 4) {
    const int ka = k0 + 2 * kh;
    v2f a, bb;
    a.x  = xb[(size_t)(ka + 0) * NNODES + m0 + m]; // A[m][k], x is K-major
    a.y  = xb[(size_t)(ka + 1) * NNODES + m0 + m];
    bb.x = Bsh[(ka + 0) * 16 + m];                 // B[k][n]
    bb.y = Bsh[(ka + 1) * 16 + m];
    acc = __builtin_amdgcn_wmma_f32_16x16x4_f32(false, a, false, bb,
                                                (short)0, acc, false, false);
  }
  float* outp = feat1 + ((size_t)b * NNODES + m0) * HID1 + j0;
  #pragma unroll
  for (int i = 0; i < 8; ++i) {
    int row = i + 8 * kh;                          // D layout: M = vgpr + 8*half
    outp[(size_t)row * HID1 + m] = acc[i];
  }
}

// -------------------- el/er for layer 1 (per node, per head) ---------------
__global__ void k_elr1(const float* __restrict__ feat1, const float* __restrict__ al1,
                       const float* __restrict__ ar1, float* __restrict__ el1,
                       float* __restrict__ er1) {
  int t = blockIdx.x * blockDim.x + threadIdx.x;   // (b*N+n)*4 + h
  if (t >= BATCHN * NNODES * 4) return;
  int h  = t & 3;
  int bn = t >> 2;
  const float* f = feat1 + (size_t)bn * HID1 + h * 8;
  float el = 0.f, er = 0.f;
  #pragma unroll
  for (int d = 0; d < 8; ++d) { el += f[d] * al1[h * 8 + d]; er += f[d] * ar1[h * 8 + d]; }
  el1[t] = el; er1[t] = er;
}

// ------------- layer-1 attention: one wave per (batch,node) ----------------
// lane j = head(j>>3), dim(j&7); online softmax over incoming edges
__global__ void k_attn1(const float* __restrict__ feat1, const float* __restrict__ el1,
                        const float* __restrict__ er1, const float* __restrict__ b1,
                        const int* __restrict__ offs, const int* __restrict__ esrc,
                        float* __restrict__ h1) {
  const int lane = threadIdx.x & 31;
  const int wid  = (blockIdx.x * blockDim.x + threadIdx.x) >> 5;
  const int b = wid >> 12, n = wid & (NNODES - 1);
  const int h = lane >> 3;
  const int s0 = offs[n], s1 = offs[n + 1];
  const float ern = er1[(((size_t)b << 12) + n) * 4 + h];
  const float* fb  = feat1 + ((size_t)b << 12) * HID1;
  const float* elb = el1 + (((size_t)b << 12) * 4);
  float mx = -1e30f, den = 0.f, acc = 0.f;
  for (int e = s0; e < s1; ++e) {
    int s = esrc[e];
    if (e + 1 < s1) __builtin_prefetch(&fb[(size_t)esrc[e + 1] * HID1], 0, 0);
    float lg = elb[s * 4 + h] + ern;
    lg = lg > 0.f ? lg : 0.2f * lg;                // LeakyReLU(0.2)
    float nm = fmaxf(mx, lg);
    float sc = __expf(mx - nm);
    float w  = __expf(lg - nm);
    den = den * sc + w;
    acc = acc * sc + w * fb[(size_t)s * HID1 + lane];  // coalesced 32f row
    mx = nm;
  }
  float v  = acc / den + b1[lane];
  float sp = __logf(1.f + __expf(v));              // softplus
  h1[(size_t)wid * HID1 + lane] = v * tanhf(sp);   // mish
}

// ------------------------- GEMM2: feat2 = h1 @ W2 --------------------------
// h1: [B][4096][32]  W2: [32][64]  feat2: [B][4096][64]
__global__ void k_gemm2(const float* __restrict__ h1, const float* __restrict__ W2,
                        float* __restrict__ feat2) {
  __shared__ float Bsh[HID1 * 16];                 // 32K x 16N slab, 2KB
  const int lane = threadIdx.x & 31;
  const int wv   = threadIdx.x >> 5;
  const int b    = blockIdx.z;
  const int j0   = blockIdx.y * 16;
  const int m0   = (blockIdx.x * 4 + wv) * 16;
  const int m    = lane & 15;
  const int kh   = lane >> 4;

  for (int idx = threadIdx.x; idx < HID1 * 16; idx += blockDim.x) {
    int k = idx >> 4, nn = idx & 15;
    Bsh[idx] = W2[k * OUTCH + j0 + nn];
  }
  __syncthreads();

  const float* hb = h1 + ((size_t)b << 12) * HID1;
  v8f acc = {};
  for (int k0 = 0; k0 < HID1; k0 += 4) {
    const int ka = k0 + 2 * kh;
    v2f a, bb;
    a.x  = hb[(size_t)(m0 + m) * HID1 + ka + 0];   // h1 is row-major in K
    a.y  = hb[(size_t)(m0 + m) * HID1 + ka + 1];
    bb.x = Bsh[(ka + 0) * 16 + m];
    bb.y = Bsh[(ka + 1) * 16 + m];
    acc = __builtin_amdgcn_wmma_f32_16x16x4_f32(false, a, false, bb,
                                                (short)0, acc, false, false);
  }
  float* outp = feat2 + ((size_t)b * NNODES + m0) * OUTCH + j0;
  #pragma unroll
  for (int i = 0; i < 8; ++i) {
    int row = i + 8 * kh;
    outp[(size_t)row * OUTCH + m] = acc[i];
  }
}

// ----------------------- el/er for layer 2 (1 head, 64d) -------------------
__global__ void k_elr2(const float* __restrict__ feat2, const float* __restrict__ al2,
                       const float* __restrict__ ar2, float* __restrict__ el2,
                       float* __restrict__ er2) {
  int t = blockIdx.x * blockDim.x + threadIdx.x;   // b*N + n
  if (t >= BATCHN * NNODES) return;
  const float* f = feat2 + (size_t)t * OUTCH;
  float el = 0.f, er = 0.f;
  #pragma unroll 8
  for (int o = 0; o < OUTCH; ++o) { el += f[o] * al2[o]; er += f[o] * ar2[o]; }
  el2[t] = el; er2[t] = er;
}

// ------------- layer-2 attention: one wave per (batch,node) ----------------
// lane owns channels {lane, lane+32}
__global__ void k_attn2(const float* __restrict__ feat2, const float* __restrict__ el2,
                        const float* __restrict__ er2, const float* __restrict__ b2,
                        const int* __restrict__ offs, const int* __restrict__ esrc,
                        float* __restrict__ h2) {
  const int lane = threadIdx.x & 31;
  const int wid  = (blockIdx.x * blockDim.x + threadIdx.x) >> 5;
  const int b = wid >> 12, n = wid & (NNODES - 1);
  const int s0 = offs[n], s1 = offs[n + 1];
  const float ern = er2[wid];
  const float* fb  = feat2 + ((size_t)b << 12) * OUTCH;
  const float* elb = el2 + ((size_t)b << 12);
  float mx = -1e30f, den = 0.f, a0 = 0.f, a1 = 0.f;
  for (int e = s0; e < s1; ++e) {
    int s = esrc[e];
    if (e + 1 < s1) __builtin_prefetch(&fb[(size_t)esrc[e + 1] * OUTCH], 0, 0);
    float lg = elb[s] + ern;
    lg = lg > 0.f ? lg : 0.2f * lg;
    float nm = fmaxf(mx, lg);
    float sc = __expf(mx - nm);
    float w  = __expf(lg - nm);
    den = den * sc + w;
    const float* fr = fb + (size_t)s * OUTCH;      // coalesced 64f row
    a0 = a0 * sc + w * fr[lane];
    a1 = a1 * sc + w * fr[lane + 32];
    mx = nm;
  }
  float* o = h2 + (size_t)wid * OUTCH;
  o[lane]      = a0 / den + b2[lane];
  o[lane + 32] = a1 / den + b2[lane + 32];
}

// ------------- transpose [B][N][64] -> [B][64][N] via LDS tiles ------------
__global__ void k_transpose(const float* __restrict__ h2, float* __restrict__ out) {
  __shared__ float t[16][17];
  const int b = blockIdx.z, o0 = blockIdx.y * 16, n0 = blockIdx.x * 16;
  const int tx = threadIdx.x, ty = threadIdx.y;
  t[ty][tx] = h2[((size_t)b * NNODES + n0 + ty) * OUTCH + o0 + tx];
  __syncthreads();
  out[((size_t)b * OUTCH + o0 + ty) * NNODES + n0 + tx] = t[tx][ty];
}

// ---------------------------------------------------------------------------
extern "C" void kernel_launch(void* const* d_in, const int* in_sizes, int n_in,
                              void* d_out, int out_size, void* d_ws, size_t ws_size,
                              hipStream_t stream) {
  const float* x   = (const float*)d_in[0];
  const float* W1  = (const float*)d_in[1];
  const float* al1 = (const float*)d_in[2];
  const float* ar1 = (const float*)d_in[3];
  const float* b1  = (const float*)d_in[4];
  const float* W2  = (const float*)d_in[5];
  const float* al2 = (const float*)d_in[6];
  const float* ar2 = (const float*)d_in[7];
  const float* b2  = (const float*)d_in[8];
  const int*   src = (const int*)d_in[9];
  const int*   dst = (const int*)d_in[10];
  const int    E   = in_sizes[9];
  float* out = (float*)d_out;

  // carve workspace
  char* w = (char*)d_ws;
  size_t off = 0;
  auto carve = [&](size_t bytes) -> void* {
    void* p = w + off;
    off += (bytes + 255) & ~(size_t)255;
    return p;
  };
  float* feat1 = (float*)carve((size_t)BATCHN * NNODES * HID1 * 4);
  float* el1   = (float*)carve((size_t)BATCHN * NNODES * 4 * 4);
  float* er1   = (float*)carve((size_t)BATCHN * NNODES * 4 * 4);
  float* h1    = (float*)carve((size_t)BATCHN * NNODES * HID1 * 4);
  float* feat2 = (float*)carve((size_t)BATCHN * NNODES * OUTCH * 4);
  float* el2   = (float*)carve((size_t)BATCHN * NNODES * 4);
  float* er2   = (float*)carve((size_t)BATCHN * NNODES * 4);
  float* h2    = (float*)carve((size_t)BATCHN * NNODES * OUTCH * 4);
  int* offs    = (int*)carve((NNODES + 1) * 4);
  int* counts  = (int*)carve(NNODES * 4);
  int* cursor  = (int*)carve(NNODES * 4);
  int* esrc    = (int*)carve((size_t)E * 4);

  const int TB = 256;
  // CSR build (dst-sorted adjacency)
  k_zero <<<(NNODES + TB - 1) / TB, TB, 0, stream>>>(counts, cursor, NNODES);
  k_count<<<(E + TB - 1) / TB, TB, 0, stream>>>(dst, counts, E);
  k_scan <<<1, 1, 0, stream>>>(counts, offs);
  k_fill <<<(E + TB - 1) / TB, TB, 0, stream>>>(src, dst, offs, cursor, esrc, E);

  // Layer 1
  k_gemm1<<<dim3(NNODES / 64, HID1 / 16, BATCHN), 128, 0, stream>>>(x, W1, feat1);
  k_elr1 <<<(BATCHN * NNODES * 4 + TB - 1) / TB, TB, 0, stream>>>(feat1, al1, ar1, el1, er1);
  k_attn1<<<(BATCHN * NNODES) / 4, 128, 0, stream>>>(feat1, el1, er1, b1, offs, esrc, h1);

  // Layer 2
  k_gemm2<<<dim3(NNODES / 64, OUTCH / 16, BATCHN), 128, 0, stream>>>(h1, W2, feat2);
  k_elr2 <<<(BATCHN * NNODES + TB - 1) / TB, TB, 0, stream>>>(feat2, al2, ar2, el2, er2);
  k_attn2<<<(BATCHN * NNODES) / 4, 128, 0, stream>>>(feat2, el2, er2, b2, offs, esrc, h2);

  // readout transpose to [B][64][H][W]
  k_transpose<<<dim3(NNODES / 16, OUTCH / 16, BATCHN), dim3(16, 16), 0, stream>>>(h2, out);
  (void)n_in; (void)out_size; (void)ws_size;
}